// GAT_56487409877509
// MI455X (gfx1250) — compile-verified
//
#include <hip/hip_runtime.h>

// ---------------- problem constants (match reference) ----------------
#define NNODES 50000
#define NEDGES 800000
#define FIN    128      // in_feats
#define FH     128      // H * HID
#define NHEADS 4
#define HIDF   32
#define NEG_SLOPE 0.2f
#define ENC_NEG_INF 0x007FFFFFu  // order-preserving encoding of -inf

typedef float v2f __attribute__((ext_vector_type(2)));
typedef float v8f __attribute__((ext_vector_type(8)));

// order-preserving float<->uint map (monotone, handles negatives)
__device__ __forceinline__ unsigned fenc(float f) {
  unsigned u = __float_as_uint(f);
  return (u & 0x80000000u) ? ~u : (u | 0x80000000u);
}
__device__ __forceinline__ float fdec(unsigned u) {
  u = (u & 0x80000000u) ? (u & 0x7FFFFFFFu) : ~u;
  return __uint_as_float(u);
}

// ---------------------------------------------------------------------
// out[M x Nc] = act(A[M x K]) @ W[K x Nc], fp32 via V_WMMA_F32_16X16X4_F32.
// grid.x = M/16; block = 256 threads = 8 wave32s; wave w owns column tile w
// (Nc == 128 == 8 * 16). One wave -> one 16x16 D tile, K-loop step 4.
// ---------------------------------------------------------------------
__global__ __launch_bounds__(256)
void gemm_wmma_f32(const float* __restrict__ A, const float* __restrict__ W,
                   float* __restrict__ out, int K, int Nc, int applyRelu) {
  const int lane = threadIdx.x & 31;
  const int wave = threadIdx.x >> 5;
  const int l16  = lane & 15;
  const int hi   = lane >> 4;           // 0: K pair {0,1}, 1: K pair {2,3}
  const size_t m0 = (size_t)blockIdx.x * 16;
  const int n0 = wave * 16;

  const float* arow = A + (m0 + (size_t)l16) * (size_t)K;
  v8f acc = {};
  for (int k = 0; k < K; k += 4) {
    const int ka = k + 2 * hi;
    // A frag: 16x4, lane l -> row (l&15), K = ka, ka+1 (contiguous float2)
    float2 av = *(const float2*)(arow + ka);
    v2f a;
    a.x = applyRelu ? fmaxf(av.x, 0.f) : av.x;
    a.y = applyRelu ? fmaxf(av.y, 0.f) : av.y;
    // B frag: 4x16, lane l -> col n0+(l&15), rows ka, ka+1
    v2f b;
    b.x = W[(size_t)ka * Nc + n0 + l16];
    b.y = W[(size_t)(ka + 1) * Nc + n0 + l16];
    acc = __builtin_amdgcn_wmma_f32_16x16x4_f32(
        /*neg_a=*/false, a, /*neg_b=*/false, b,
        /*c_mod=*/(short)0, acc, /*reuse_a=*/false, /*reuse_b=*/false);
  }
  // D layout: VGPR r -> row m0 + r + 8*hi, col n0 + (l&15)
#pragma unroll
  for (int r = 0; r < 8; ++r)
    out[(m0 + r + 8 * hi) * (size_t)Nc + n0 + l16] = acc[r];
}

// init aggregation buffer with bias broadcast; init segment-max/-sum buffers
__global__ void init_agg(float* __restrict__ aggOut, const float* __restrict__ bias,
                         unsigned* __restrict__ mEnc, float* __restrict__ denom) {
  size_t idx = (size_t)blockIdx.x * blockDim.x + threadIdx.x;
  if (idx < (size_t)NNODES * FH) aggOut[idx] = bias[idx & (FH - 1)];
  if (idx < (size_t)NNODES * NHEADS) { mEnc[idx] = ENC_NEG_INF; denom[idx] = 0.f; }
}

// el[n,h] = <feat[n,h,:], al[h,:]>, er likewise
__global__ void node_attn(const float* __restrict__ feat, const float* __restrict__ al,
                          const float* __restrict__ ar, float* __restrict__ el,
                          float* __restrict__ er) {
  int idx = blockIdx.x * blockDim.x + threadIdx.x;
  if (idx >= NNODES * NHEADS) return;
  const int h = idx & (NHEADS - 1);
  const int n = idx >> 2;
  const float* f   = feat + (size_t)n * FH + h * HIDF;
  const float* alh = al + h * HIDF;
  const float* arh = ar + h * HIDF;
  float sl = 0.f, sr = 0.f;
#pragma unroll
  for (int k = 0; k < HIDF; ++k) { float v = f[k]; sl += v * alh[k]; sr += v * arh[k]; }
  el[idx] = sl; er[idx] = sr;
}

// pass A: e = leaky_relu(el[src]+er[dst]); segment max into mEnc[dst]
__global__ void edge_max(const int* __restrict__ src, const int* __restrict__ dst,
                         const float* __restrict__ el, const float* __restrict__ er,
                         float* __restrict__ ebuf, unsigned* __restrict__ mEnc) {
  int idx = blockIdx.x * blockDim.x + threadIdx.x;
  if (idx >= NEDGES * NHEADS) return;
  const int h = idx & (NHEADS - 1);
  const int e = idx >> 2;
  const int s = src[e], d = dst[e];
  float v = el[s * NHEADS + h] + er[d * NHEADS + h];
  v = (v > 0.f) ? v : NEG_SLOPE * v;
  ebuf[idx] = v;
  atomicMax(mEnc + d * NHEADS + h, fenc(v));
}

// pass B: ex = exp(e - m[dst]); segment sum into denom[dst]
__global__ void edge_exp(const int* __restrict__ dst, const unsigned* __restrict__ mEnc,
                         float* __restrict__ ebuf, float* __restrict__ denom) {
  int idx = blockIdx.x * blockDim.x + threadIdx.x;
  if (idx >= NEDGES * NHEADS) return;
  const int h = idx & (NHEADS - 1);
  const int e = idx >> 2;
  const int d = dst[e];
  float ex = expf(ebuf[idx] - fdec(mEnc[d * NHEADS + h]));
  ebuf[idx] = ex;
  atomicAdd(denom + d * NHEADS + h, ex);
}

// pass C: agg[dst,h,:] += feat[src,h,:] * (ex / denom[dst,h])
// 32 threads per edge; each thread: one eighth of a head (float4 of features)
__global__ void edge_agg(const int* __restrict__ src, const int* __restrict__ dst,
                         const float* __restrict__ feat, const float* __restrict__ ebuf,
                         const float* __restrict__ denom, float* __restrict__ aggOut) {
  int idx = blockIdx.x * blockDim.x + threadIdx.x;   // E * 32 threads
  if (idx >= NEDGES * 32) return;
  const int e = idx >> 5;
  const int t = idx & 31;
  const int h = t >> 3;          // head
  const int q = t & 7;           // which 4-float chunk of the head
  const int s = src[e], d = dst[e];
  const float alpha = ebuf[e * NHEADS + h] / denom[d * NHEADS + h];
  const int fo = h * HIDF + q * 4;
  const float4 v = *(const float4*)(feat + (size_t)s * FH + fo);
  float* o = aggOut + (size_t)d * FH + fo;
  atomicAdd(o + 0, v.x * alpha);
  atomicAdd(o + 1, v.y * alpha);
  atomicAdd(o + 2, v.z * alpha);
  atomicAdd(o + 3, v.w * alpha);
}

// out[n] = mean_h(agg[n,h,:]) @ fcW + fcb
__global__ void final_fc(const float* __restrict__ agg, const float* __restrict__ fcW,
                         const float* __restrict__ fcb, float* __restrict__ out) {
  int n = blockIdx.x * blockDim.x + threadIdx.x;
  if (n >= NNODES) return;
  const float* f = agg + (size_t)n * FH;
  float acc = 0.f;
#pragma unroll
  for (int h = 0; h < NHEADS; ++h)
#pragma unroll
    for (int k = 0; k < HIDF; ++k)
      acc += f[h * HIDF + k] * fcW[k];
  out[n] = acc * (1.0f / NHEADS) + fcb[0];
}

extern "C" void kernel_launch(void* const* d_in, const int* in_sizes, int n_in,
                              void* d_out, int out_size, void* d_ws, size_t ws_size,
                              hipStream_t stream) {
  const float* x   = (const float*)d_in[0];
  const int*   src = (const int*)d_in[1];
  const int*   dst = (const int*)d_in[2];
  const float* W1  = (const float*)d_in[3];
  const float* al1 = (const float*)d_in[4];
  const float* ar1 = (const float*)d_in[5];
  const float* b1  = (const float*)d_in[6];
  const float* W2  = (const float*)d_in[7];
  const float* al2 = (const float*)d_in[8];
  const float* ar2 = (const float*)d_in[9];
  const float* b2  = (const float*)d_in[10];
  const float* fcW = (const float*)d_in[11];
  const float* fcb = (const float*)d_in[12];
  float* out = (float*)d_out;

  // workspace layout (all fits in ~68 MB; entire working set is L2-resident)
  float*    buf0  = (float*)d_ws;                          // feat   [N,128]
  float*    buf1  = buf0 + (size_t)NNODES * FH;            // agg/h  [N,128]
  float*    el    = buf1 + (size_t)NNODES * FH;            // [N,H]
  float*    er    = el + (size_t)NNODES * NHEADS;          // [N,H]
  unsigned* mEnc  = (unsigned*)(er + (size_t)NNODES * NHEADS); // [N,H]
  float*    denom = (float*)(mEnc + (size_t)NNODES * NHEADS);  // [N,H]
  float*    ebuf  = denom + (size_t)NNODES * NHEADS;       // [E,H]

  const int TPB = 256;
  const int gInit = (int)(((size_t)NNODES * FH + TPB - 1) / TPB);
  const int gNode = (NNODES * NHEADS + TPB - 1) / TPB;
  const int gEH   = (NEDGES * NHEADS + TPB - 1) / TPB;
  const int gE32  = (NEDGES * 32 + TPB - 1) / TPB;
  const int gGemm = NNODES / 16;   // 3125, exact
  const int gFc   = (NNODES + TPB - 1) / TPB;

  // ---------------- layer 1 ----------------
  gemm_wmma_f32<<<gGemm, TPB, 0, stream>>>(x, W1, buf0, FIN, FH, 0);
  init_agg<<<gInit, TPB, 0, stream>>>(buf1, b1, mEnc, denom);
  node_attn<<<gNode, TPB, 0, stream>>>(buf0, al1, ar1, el, er);
  edge_max<<<gEH, TPB, 0, stream>>>(src, dst, el, er, ebuf, mEnc);
  edge_exp<<<gEH, TPB, 0, stream>>>(dst, mEnc, ebuf, denom);
  edge_agg<<<gE32, TPB, 0, stream>>>(src, dst, buf0, ebuf, denom, buf1);
  // ---------------- layer 2 (ReLU folded into GEMM A-load) ----------------
  gemm_wmma_f32<<<gGemm, TPB, 0, stream>>>(buf1, W2, buf0, FH, FH, 1);
  init_agg<<<gInit, TPB, 0, stream>>>(buf1, b2, mEnc, denom);
  node_attn<<<gNode, TPB, 0, stream>>>(buf0, al2, ar2, el, er);
  edge_max<<<gEH, TPB, 0, stream>>>(src, dst, el, er, ebuf, mEnc);
  edge_exp<<<gEH, TPB, 0, stream>>>(dst, mEnc, ebuf, denom);
  edge_agg<<<gE32, TPB, 0, stream>>>(src, dst, buf0, ebuf, denom, buf1);
  // ---------------- head-mean + fc ----------------
  final_fc<<<gFc, TPB, 0, stream>>>(buf1, fcW, fcb, out);
}